// ScalarAttentionStrategy_38250978738512
// MI455X (gfx1250) — compile-verified
//
#include <hip/hip_runtime.h>
#include <math.h>

// Problem constants (from the reference)
#define D        64
#define L        50
#define PAD_IDX  0

#define WAVES_PER_BLOCK 4
#define BLOCK_THREADS   (WAVES_PER_BLOCK * 32)
#define NPAIR           (L / 2)   // 25 token pairs

#ifndef __has_builtin
#define __has_builtin(x) 0
#endif

#if defined(__HIP_DEVICE_COMPILE__) && __has_builtin(__builtin_amdgcn_global_load_async_to_lds_b128)
#define USE_ASYNC_LDS 1
#else
#define USE_ASYNC_LDS 0
#endif

// 16-byte int vector matching the builtin's pointee types (AS1 src, AS3 dst).
typedef int v4i __attribute__((vector_size(16)));
typedef __attribute__((address_space(1))) v4i g_v4i;
typedef __attribute__((address_space(3))) v4i l_v4i;

__global__ __launch_bounds__(BLOCK_THREADS)
void emb_attn_pool_kernel(const int* __restrict__ idx,
                          const float* __restrict__ emb,
                          const float* __restrict__ attn_w,
                          const float* __restrict__ attn_b,
                          float* __restrict__ out,
                          int rows)
{
    // Per-wave staging: 50 tokens x 64 f32 = 12.8 KB, x4 waves = 51.2 KB
    __shared__ float lds_emb[WAVES_PER_BLOCK][L * D];

    const int lane = threadIdx.x & 31;
    const int wv   = threadIdx.x >> 5;
    const int row  = blockIdx.x * WAVES_PER_BLOCK + wv;
    if (row >= rows) return;

    // Wave-uniform row id -> scalar (s_load) index reads through the constant cache.
    const int  urow    = __builtin_amdgcn_readfirstlane(row);
    const int* idx_row = idx + urow * L;

    const int half = lane >> 4;      // 0: even token of pair, 1: odd token
    const int sub  = lane & 15;      // 16B chunk / float4 slot within a token row

    // ---- Async gather into LDS: one b128 instruction moves TWO embedding rows
    // (lanes 0-15 -> token 2p, lanes 16-31 -> token 2p+1; 16 B per lane).
    // 25 outstanding ASYNCcnt requests per wave hide HBM gather latency.
#pragma unroll
    for (int p = 0; p < NPAIR; ++p) {
        const int tokA = idx_row[2 * p];
        const int tokB = idx_row[2 * p + 1];
        const int tok  = (lane < 16) ? tokA : tokB;
        const float* gp = emb + (size_t)(unsigned)tok * D + sub * 4;
        float*       lp = &lds_emb[wv][(2 * p + half) * D + sub * 4];
#if USE_ASYNC_LDS
        __builtin_amdgcn_global_load_async_to_lds_b128(
            (g_v4i*)gp, (l_v4i*)lp, /*offset=*/0, /*cpol=*/0);
#else
        float4 v = *(const float4*)gp;
        *(float4*)lp = v;
#endif
    }
#if USE_ASYNC_LDS
#if __has_builtin(__builtin_amdgcn_s_wait_asynccnt)
    __builtin_amdgcn_s_wait_asynccnt(0);
    asm volatile("" ::: "memory");
#else
    asm volatile("s_wait_asynccnt 0" ::: "memory");
#endif
#else
    asm volatile("s_wait_dscnt 0" ::: "memory");
#endif

    // ---- Two parallel online-softmax states: half 0 handles even tokens,
    // half 1 handles odd tokens. Each half holds all 64 dims (4 per lane).
    const float4 w4 = ((const float4*)attn_w)[sub];
    const float  bs = attn_b[0];
    const float  NEG_INF = -__builtin_inff();

    float m  = NEG_INF;
    float ss = 0.0f;
    float ax = 0.0f, ay = 0.0f, az = 0.0f, aw = 0.0f;

#pragma unroll
    for (int p = 0; p < NPAIR; ++p) {
        const int tokA = idx_row[2 * p];
        const int tokB = idx_row[2 * p + 1];
        const int tok  = (lane < 16) ? tokA : tokB;     // half-uniform
        const bool valid = (tok != PAD_IDX);

        const float4 e = *(const float4*)&lds_emb[wv][(2 * p + half) * D + sub * 4];
        float s = fmaf(e.x, w4.x, fmaf(e.y, w4.y, fmaf(e.z, w4.z, e.w * w4.w)));
        // 4-step butterfly within each 16-lane half (xor<=8 never crosses halves)
        s += __shfl_xor(s, 8, 32);
        s += __shfl_xor(s, 4, 32);
        s += __shfl_xor(s, 2, 32);
        s += __shfl_xor(s, 1, 32);

        const float sc   = valid ? (s + bs) : NEG_INF;
        const float mn   = fmaxf(m, sc);
        // guard: before any valid token m == -inf; never feed (-inf - -inf) to exp
        const float corr = (m == NEG_INF) ? 0.0f : __expf(m - mn);
        const float pw   = valid ? __expf(sc - mn) : 0.0f;
        ss = fmaf(ss, corr, pw);
        ax = fmaf(ax, corr, pw * e.x);
        ay = fmaf(ay, corr, pw * e.y);
        az = fmaf(az, corr, pw * e.z);
        aw = fmaf(aw, corr, pw * e.w);
        m  = mn;
    }

    // ---- Merge the two half-wave softmax states (lane l <-> lane l^16 hold
    // the same dims, so a single xor-16 exchange aligns everything).
    const float om  = __shfl_xor(m,  16, 32);
    const float oss = __shfl_xor(ss, 16, 32);
    const float oax = __shfl_xor(ax, 16, 32);
    const float oay = __shfl_xor(ay, 16, 32);
    const float oaz = __shfl_xor(az, 16, 32);
    const float oaw = __shfl_xor(aw, 16, 32);

    const float M  = fmaxf(m, om);
    const float c0 = (m  == NEG_INF) ? 0.0f : __expf(m  - M);
    const float c1 = (om == NEG_INF) ? 0.0f : __expf(om - M);

    const float S   = ss * c0 + oss * c1;
    const float inv = (S > 0.0f) ? (1.0f / S) : 0.0f;  // all-PAD row -> zeros

    float4 r;
    r.x = (ax * c0 + oax * c1) * inv;
    r.y = (ay * c0 + oay * c1) * inv;
    r.z = (az * c0 + oaz * c1) * inv;
    r.w = (aw * c0 + oaw * c1) * inv;

    if (lane < 16)
        ((float4*)out)[urow * (D / 4) + sub] = r;
}

extern "C" void kernel_launch(void* const* d_in, const int* in_sizes, int n_in,
                              void* d_out, int out_size, void* d_ws, size_t ws_size,
                              hipStream_t stream)
{
    const int*   idx  = (const int*)d_in[0];
    const float* emb  = (const float*)d_in[1];
    const float* w    = (const float*)d_in[2];
    const float* bias = (const float*)d_in[3];
    float*       out  = (float*)d_out;

    const int rows = in_sizes[0] / L;  // B = 16384
    dim3 grid((rows + WAVES_PER_BLOCK - 1) / WAVES_PER_BLOCK);
    emb_attn_pool_kernel<<<grid, BLOCK_THREADS, 0, stream>>>(idx, emb, w, bias, out, rows);
}